// GraphNet_80522046866010
// MI455X (gfx1250) — compile-verified
//
#include <hip/hip_runtime.h>
#include <hip/hip_bf16.h>

// GraphSAGE 3-layer GNN for MI455X (gfx1250, wave32).
//  - degree sums computed once; 1/deg folded into the edge scatter so agg holds
//    the weighted MEAN directly.
//  - edge scatter: wave-per-edge float4 gather (x table = 25.6MB, L2-resident)
//    + global_atomic_add_f32.
//  - linear layers: fp32 WMMA (v_wmma_f32_16x16x4_f32). W staged in LDS
//    (bank-conflict-free padded pitch), A (global) + B (LDS) double-buffered so
//    the WMMA chain doesn't sit behind s_wait_loadcnt 0 every instruction.

typedef float v2f __attribute__((ext_vector_type(2)));
typedef float v8f __attribute__((ext_vector_type(8)));

#define NN 50000      // nodes (exactly 3125 * 16)
#define DD 128        // per-half input feature dim (all layers)
#define DD2 256       // concat dim
#define WPITCH 260    // LDS row pitch in floats: 16B-aligned rows, 4-bank skew

// ---------------------------------------------------------------- utilities
__global__ void zero_f32_kernel(float* __restrict__ p, int n4) {
    int i = blockIdx.x * blockDim.x + threadIdx.x;
    if (i < n4) ((float4*)p)[i] = make_float4(0.f, 0.f, 0.f, 0.f);
}

__global__ void deg_kernel(const int* __restrict__ row,
                           const float* __restrict__ ew,
                           float* __restrict__ deg, int E) {
    int e = blockIdx.x * blockDim.x + threadIdx.x;
    if (e < E) atomicAdd(&deg[row[e]], ew[e]);
}

__global__ void rdeg_kernel(const float* __restrict__ deg,
                            float* __restrict__ rdeg, int n) {
    int i = blockIdx.x * blockDim.x + threadIdx.x;
    if (i < n) rdeg[i] = 1.0f / fmaxf(deg[i], 1.0f);
}

// ------------------------------------------------------------- edge scatter
// One wave per edge; lane l moves features [4l,4l+4). The edge weight is
// pre-multiplied by 1/deg(dst) so agg accumulates the weighted mean directly.
__global__ void scatter_kernel(const float* __restrict__ x,
                               const int* __restrict__ row,
                               const int* __restrict__ col,
                               const float* __restrict__ ew,
                               const float* __restrict__ rdeg,
                               float* __restrict__ agg, int E) {
    int t = blockIdx.x * blockDim.x + threadIdx.x;
    int e = t >> 5;
    int lane = t & 31;
    if (e >= E) return;
    int r = row[e];
    int c = col[e];
    float w = ew[e] * rdeg[r];
    float4 v = ((const float4*)(x + (size_t)c * DD))[lane];
    float* dst = agg + (size_t)r * DD + lane * 4;
    atomicAdd(dst + 0, v.x * w);
    atomicAdd(dst + 1, v.y * w);
    atomicAdd(dst + 2, v.z * w);
    atomicAdd(dst + 3, v.w * w);
}

// ---------------------------------------------------------- WMMA linear layer
// OUT[N, DOUT] = act( [X | AGG] @ W^T + bias ),  W is [DOUT, 256] row-major.
// One wave owns a 16-row strip and ALL DOUT columns (DOUT/16 accumulators).
// W lives in LDS (copied once per block); A and B fragments are pipelined one
// K-step ahead so DOUT/16 WMMAs issue back-to-back per iteration.
template <int DOUT, bool RELU>
__global__ void __launch_bounds__(256)
sage_linear_wmma(const float* __restrict__ X,
                 const float* __restrict__ AGG,
                 const float* __restrict__ W,
                 const float* __restrict__ BIAS,
                 float* __restrict__ OUT) {
    constexpr int NT = DOUT / 16;                 // output column tiles / wave
    __shared__ __align__(16) float wlds[DOUT * WPITCH];

    // ---- stage W into LDS: DOUT rows of 256 floats, padded pitch 260.
    for (int idx = threadIdx.x; idx < DOUT * (DD2 / 4); idx += 256) {
        const int o  = idx >> 6;                  // row (output column)
        const int c4 = idx & 63;                  // float4 within row
        float4 v = ((const float4*)(W + (size_t)o * DD2))[c4];
        ((float4*)(wlds + o * WPITCH))[c4] = v;   // rows 16B-aligned (1040B pitch)
    }
    __syncthreads();

    const int wave = (blockIdx.x * blockDim.x + threadIdx.x) >> 5;
    if (wave >= NN / 16) return;                  // wave-uniform guard
    const int lane = threadIdx.x & 31;
    const int m  = lane & 15;                     // A row / B column in tile
    const int kh = lane >> 4;                     // which K-pair this lane holds
    const int row = wave * 16 + m;

    const float* __restrict__ xr = X   + (size_t)row * DD;
    const float* __restrict__ ar = AGG + (size_t)row * DD - DD;  // virtual concat

    v8f acc[NT] = {};
    v2f a_cur, b_cur[NT], a_nxt, b_nxt[NT];

    // prologue: K-step 0
    a_cur = *(const v2f*)(xr + 2 * kh);
#pragma unroll
    for (int ct = 0; ct < NT; ++ct)
        b_cur[ct] = *(const v2f*)(wlds + (ct * 16 + m) * WPITCH + 2 * kh);

#pragma unroll 4
    for (int kk = 0; kk < DD2 - 4; kk += 4) {
        const int kn = kk + 4;
        const int k  = kn + 2 * kh;
        // issue next-iteration loads first (A from global/L2, B from LDS)
        const float* __restrict__ base = (kn < DD) ? xr : ar;
        a_nxt = *(const v2f*)(base + k);
#pragma unroll
        for (int ct = 0; ct < NT; ++ct)
            b_nxt[ct] = *(const v2f*)(wlds + (ct * 16 + m) * WPITCH + k);
        // compute on current fragments: NT back-to-back WMMAs
#pragma unroll
        for (int ct = 0; ct < NT; ++ct)
            acc[ct] = __builtin_amdgcn_wmma_f32_16x16x4_f32(
                false, a_cur, false, b_cur[ct], (short)0, acc[ct], false, false);
        a_cur = a_nxt;
#pragma unroll
        for (int ct = 0; ct < NT; ++ct) b_cur[ct] = b_nxt[ct];
    }
    // epilogue K-step
#pragma unroll
    for (int ct = 0; ct < NT; ++ct)
        acc[ct] = __builtin_amdgcn_wmma_f32_16x16x4_f32(
            false, a_cur, false, b_cur[ct], (short)0, acc[ct], false, false);

    // bias (+ ReLU) and store; column fixed per lane across j.
#pragma unroll
    for (int ct = 0; ct < NT; ++ct) {
        const int colc = ct * 16 + m;
        const float bv = BIAS[colc];
#pragma unroll
        for (int j = 0; j < 8; ++j) {
            float v = acc[ct][j] + bv;
            if (RELU) v = fmaxf(v, 0.0f);
            OUT[(size_t)(wave * 16 + j + 8 * kh) * DOUT + colc] = v;
        }
    }
}

// ------------------------------------------------------------------- driver
extern "C" void kernel_launch(void* const* d_in, const int* in_sizes, int n_in,
                              void* d_out, int out_size, void* d_ws, size_t ws_size,
                              hipStream_t stream) {
    const float* x  = (const float*)d_in[0];
    const int*   ei = (const int*)d_in[1];
    const float* ew = (const float*)d_in[2];
    const float* W0 = (const float*)d_in[3];
    const float* b0 = (const float*)d_in[4];
    const float* W1 = (const float*)d_in[5];
    const float* b1 = (const float*)d_in[6];
    const float* W2 = (const float*)d_in[7];
    const float* b2 = (const float*)d_in[8];
    float* out = (float*)d_out;

    const int E = in_sizes[2];              // edge_weight count
    const int* rowI = ei;                   // edge_index[0] (dst)
    const int* colI = ei + E;               // edge_index[1] (src)

    // workspace (floats): deg[N] | rdeg[N] | agg[N*128] | hA[N*128] | hB[N*128]
    float* ws   = (float*)d_ws;
    float* deg  = ws;
    float* rdeg = ws + NN;
    float* agg  = ws + 2 * (size_t)NN;
    float* hA   = agg + (size_t)NN * DD;
    float* hB   = hA  + (size_t)NN * DD;

    const int B256 = 256;
    const int degZeroBlocks = (NN / 4 + B256 - 1) / B256;
    const int aggZeroBlocks = ((NN * DD) / 4 + B256 - 1) / B256;
    const int degBlocks     = (E + B256 - 1) / B256;
    const int rdegBlocks    = (NN + B256 - 1) / B256;
    const int scatterBlocks = ((E * 32) + B256 - 1) / B256;
    const int gemmBlocks    = ((NN / 16) + 7) / 8;          // 8 waves / block

    // degree normalization: computed once, reused by all three layers
    zero_f32_kernel<<<degZeroBlocks, B256, 0, stream>>>(deg, NN / 4);
    deg_kernel<<<degBlocks, B256, 0, stream>>>(rowI, ew, deg, E);
    rdeg_kernel<<<rdegBlocks, B256, 0, stream>>>(deg, rdeg, NN);

    // ---- layer 0: x -> hA (ReLU)
    zero_f32_kernel<<<aggZeroBlocks, B256, 0, stream>>>(agg, (NN * DD) / 4);
    scatter_kernel<<<scatterBlocks, B256, 0, stream>>>(x, rowI, colI, ew, rdeg, agg, E);
    sage_linear_wmma<128, true><<<gemmBlocks, B256, 0, stream>>>(x, agg, W0, b0, hA);

    // ---- layer 1: hA -> hB (ReLU)
    zero_f32_kernel<<<aggZeroBlocks, B256, 0, stream>>>(agg, (NN * DD) / 4);
    scatter_kernel<<<scatterBlocks, B256, 0, stream>>>(hA, rowI, colI, ew, rdeg, agg, E);
    sage_linear_wmma<128, true><<<gemmBlocks, B256, 0, stream>>>(hA, agg, W1, b1, hB);

    // ---- layer 2: hB -> out (no activation), DOUT = 64
    zero_f32_kernel<<<aggZeroBlocks, B256, 0, stream>>>(agg, (NN * DD) / 4);
    scatter_kernel<<<scatterBlocks, B256, 0, stream>>>(hB, rowI, colI, ew, rdeg, agg, E);
    sage_linear_wmma<64, false><<<gemmBlocks, B256, 0, stream>>>(hB, agg, W2, b2, out);
}